// ShuffleNetUnitNew_42331197670125
// MI455X (gfx1250) — compile-verified
//
#include <hip/hip_runtime.h>
#include <hip/hip_bf16.h>

typedef float v2f __attribute__((ext_vector_type(2)));
typedef float v8f __attribute__((ext_vector_type(8)));

#define GROUPS 3
#define BATCH 32
#define CIN   240
#define HH    56
#define WW    56
#define MID   60
#define COUT  240
#define TH    8                 // output rows per workgroup tile
#define PTS_H ((TH + 2) * WW)   // 560 halo points
#define PTS_O (TH * WW)         // 448 output points
#define SW_STRIDE1 244          // W1pad row stride (words), even, conflict-free
#define SW_STRIDE2 68           // W3d  row stride (words), even, conflict-free
#define EPS 1e-5f

__device__ __forceinline__ v8f wmma_f32(v2f a, v2f b, v8f c) {
    return __builtin_amdgcn_wmma_f32_16x16x4_f32(false, a, false, b, (short)0, c,
                                                 false, false);
}

__global__ __launch_bounds__(256) void shufflenet_fused_kernel(
    const float* __restrict__ x,
    const float* __restrict__ w1, const float* __restrict__ g1,
    const float* __restrict__ b1, const float* __restrict__ m1,
    const float* __restrict__ v1,
    const float* __restrict__ w2, const float* __restrict__ g2,
    const float* __restrict__ b2, const float* __restrict__ m2,
    const float* __restrict__ v2,
    const float* __restrict__ w3, const float* __restrict__ g3,
    const float* __restrict__ b3, const float* __restrict__ m3,
    const float* __restrict__ v3,
    float* __restrict__ out) {

    // ---- LDS (319.5 KB total, one WG per WGP) ----
    __shared__ __align__(16) float sW[16320];          // W1pad [64 x 244] then W3d [240 x 68]
    __shared__ __align__(16) float sMidH[MID * PTS_H]; // relu(bn1(conv1)) halo tile [60 x 560]
    __shared__ __align__(16) float sMid2[64 * PTS_O];  // bn2(dwconv) [64 x 448] (rows 60..63 = 0)
    __shared__ float inv1s[64], add1s[64];
    __shared__ float inv2s[64], add2s[64];
    __shared__ float inv3s[COUT], add3s[COUT];
    __shared__ float w2s[544];

    const int tid  = threadIdx.x;
    const int lane = tid & 31;
    const int wave = tid >> 5;
    const int wg   = blockIdx.x;
    const int bimg = wg / (HH / TH);        // batch image
    const int r0   = (wg % (HH / TH)) * TH; // first output row

    // ================= Phase 0: weights / BN folding into LDS =================
    if (tid < 64) {
        float iv = 0.f, ad = 0.f, iv2 = 0.f, ad2 = 0.f;
        if (tid < MID) {
            float inv = g1[tid] * rsqrtf(v1[tid] + EPS);
            iv = inv; ad = b1[tid] - m1[tid] * inv;
            float invb = g2[tid] * rsqrtf(v2[tid] + EPS);
            iv2 = invb; ad2 = b2[tid] - m2[tid] * invb;
        }
        inv1s[tid] = iv;  add1s[tid] = ad;
        inv2s[tid] = iv2; add2s[tid] = ad2;
    }
    if (tid < COUT) {
        float inv = g3[tid] * rsqrtf(v3[tid] + EPS);
        inv3s[tid] = inv; add3s[tid] = b3[tid] - m3[tid] * inv;
    }
    for (int idx = tid; idx < MID * 9; idx += 256) w2s[idx] = w2[idx];
    // Block-diagonal W1pad[64][240]: groups become a dense GEMM.
    for (int idx = tid; idx < 64 * CIN; idx += 256) {
        int m = idx / CIN, k = idx - m * CIN;
        float val = 0.f;
        if (m < MID) {
            int g = m / (MID / GROUPS);
            int kb = g * (CIN / GROUPS);
            if (k >= kb && k < kb + CIN / GROUPS) val = w1[m * (CIN / GROUPS) + (k - kb)];
        }
        sW[m * SW_STRIDE1 + k] = val;
    }
    __syncthreads();

    // ================= Phase 1: GEMM1 (WMMA f32 16x16x4) + BN1 + ReLU ========
    // mid[64 x 560] = W1pad[64 x 240] * x[240 x 560]; B streamed from HBM.
    // Halo rows outside the image are addressed CLAMPED (always in-bounds) and
    // loaded unconditionally: WMMA column n depends only on lane n's B data,
    // and the epilogue stores 0 for those points, so the garbage never escapes.
    {
        const int n    = lane & 15;
        const int half = lane >> 4;
        for (int nt = wave; nt < PTS_H / 16; nt += 8) {   // 35 N-tiles over 8 waves
            const int pt   = nt * 16 + n;
            const int hrow = pt / WW;
            const int col  = pt - hrow * WW;
            const int r    = r0 - 1 + hrow;
            const bool rv  = (r >= 0) && (r < HH);
            const int  rc  = r < 0 ? 0 : (r > HH - 1 ? HH - 1 : r);
            const float* xb = x + ((bimg * CIN) * HH + rc) * WW + col;
            v8f acc[4];
            acc[0] = v8f{}; acc[1] = v8f{}; acc[2] = v8f{}; acc[3] = v8f{};
#pragma unroll 4
            for (int k0 = 0; k0 < CIN; k0 += 4) {
                const int k = k0 + 2 * half;
                v2f bf;
                bf.x = xb[k * (HH * WW)];          // unconditional, clamped row
                bf.y = xb[(k + 1) * (HH * WW)];
#pragma unroll
                for (int mt = 0; mt < 4; ++mt) {
                    v2f af;
                    const int row = mt * 16 + n;
                    af.x = sW[row * SW_STRIDE1 + k];
                    af.y = sW[row * SW_STRIDE1 + k + 1];
                    acc[mt] = wmma_f32(af, bf, acc[mt]);
                }
            }
            // Fused BN1 + ReLU; out-of-image halo rows become conv zero-padding.
#pragma unroll
            for (int mt = 0; mt < 4; ++mt) {
#pragma unroll
                for (int v = 0; v < 8; ++v) {
                    const int m = mt * 16 + v + 8 * half;
                    if (m < MID) {
                        float val = acc[mt][v] * inv1s[m] + add1s[m];
                        val = val > 0.f ? val : 0.f;
                        sMidH[m * PTS_H + pt] = rv ? val : 0.f;
                    }
                }
            }
        }
    }
    __syncthreads();

    // ====== Phase 2a: W3 with channel-shuffle folded in (reuse sW region) ====
    for (int idx = tid; idx < COUT * 64; idx += 256) {
        int o = idx >> 6, c = idx & 63;
        float val = 0.f;
        if (c < MID) {
            int s = (c % 20) * GROUPS + c / 20;    // shuffled position of mid channel c
            if (s / 20 == o / (COUT / GROUPS)) val = w3[o * 20 + (s % 20)];
        }
        sW[o * SW_STRIDE2 + c] = val;
    }
    // ====== Phase 2b: depthwise 3x3 + BN2, entirely in LDS ===================
    for (int idx = tid; idx < 64 * PTS_O; idx += 256) {
        const int m  = idx / PTS_O;
        const int pt = idx - m * PTS_O;
        float val = 0.f;
        if (m < MID) {
            const int rr = pt / WW, cc = pt - (pt / WW) * WW;
            const float* mh = &sMidH[m * PTS_H + rr * WW + cc];
            const float* wv = &w2s[m * 9];
            float s = 0.f;
#pragma unroll
            for (int dy = 0; dy < 3; ++dy) {
#pragma unroll
                for (int dx = 0; dx < 3; ++dx) {
                    const int c2 = cc + dx - 1;
                    if (c2 >= 0 && c2 < WW) s += mh[dy * WW + dx - 1] * wv[dy * 3 + dx];
                }
            }
            val = s * inv2s[m] + add2s[m];
        }
        sMid2[m * PTS_O + pt] = val;   // rows 60..63 zeroed (pad K for GEMM2)
    }
    __syncthreads();

    // ====== Phase 3: GEMM2 (WMMA) + BN3 + ReLU + residual ====================
    // out[240 x 448] = W3d[240 x 64] * mid2[64 x 448]
    {
        const int n    = lane & 15;
        const int half = lane >> 4;
        const int NT   = PTS_O / 16;              // 28
        for (int p = wave; p < 15 * NT; p += 8) { // 15 M-tiles x 28 N-tiles
            const int mt = p / NT, nt = p - mt * NT;
            const int pt = nt * 16 + n;
            v8f acc = v8f{};
#pragma unroll 4
            for (int k0 = 0; k0 < 64; k0 += 4) {
                const int k = k0 + 2 * half;
                v2f bf;
                bf.x = sMid2[k * PTS_O + pt];
                bf.y = sMid2[(k + 1) * PTS_O + pt];
                v2f af;
                const int row = mt * 16 + n;
                af.x = sW[row * SW_STRIDE2 + k];
                af.y = sW[row * SW_STRIDE2 + k + 1];
                acc = wmma_f32(af, bf, acc);
            }
            const int rr = pt / WW, cc = pt - (pt / WW) * WW;
            const int r  = r0 + rr;
#pragma unroll
            for (int v = 0; v < 8; ++v) {
                const int o = mt * 16 + v + 8 * half;
                const int gi = ((bimg * COUT + o) * HH + r) * WW + cc;
                float val = acc[v] * inv3s[o] + add3s[o];
                val = val > 0.f ? val : 0.f;
                out[gi] = val + x[gi];
            }
        }
    }
}

extern "C" void kernel_launch(void* const* d_in, const int* in_sizes, int n_in,
                              void* d_out, int out_size, void* d_ws, size_t ws_size,
                              hipStream_t stream) {
    (void)in_sizes; (void)n_in; (void)d_ws; (void)ws_size; (void)out_size;
    const float* x  = (const float*)d_in[0];
    const float* w1 = (const float*)d_in[1];
    const float* g1 = (const float*)d_in[2];
    const float* b1 = (const float*)d_in[3];
    const float* m1 = (const float*)d_in[4];
    const float* v1 = (const float*)d_in[5];
    const float* w2 = (const float*)d_in[6];
    const float* g2 = (const float*)d_in[7];
    const float* b2 = (const float*)d_in[8];
    const float* m2 = (const float*)d_in[9];
    const float* v2 = (const float*)d_in[10];
    const float* w3 = (const float*)d_in[11];
    const float* g3 = (const float*)d_in[12];
    const float* b3 = (const float*)d_in[13];
    const float* m3 = (const float*)d_in[14];
    const float* v3 = (const float*)d_in[15];
    float* out = (float*)d_out;

    dim3 grid(BATCH * (HH / TH));   // 224 workgroups
    dim3 block(256);                // 8 wave32 per WGP
    shufflenet_fused_kernel<<<grid, block, 0, stream>>>(
        x, w1, g1, b1, m1, v1, w2, g2, b2, m2, v2, w3, g3, b3, m3, v3, out);
}